// DigitCapsules_90391881712309
// MI455X (gfx1250) — compile-verified
//
#include <hip/hip_runtime.h>

// ---------------------------------------------------------------------------
// Capsule dynamic routing for MI455X (gfx1250, wave32, WMMA).
//   B=32 batches, R=576 routes, K=576 capsules, D=16, C_in=8, 3 routing iters.
// Logits are rank-16 (b = U * Wacc^T, Wacc = running sum of v), so routing is
// pure f32 GEMM work on V_WMMA_F32_16X16X4_F32, LDS-resident per batch.
// All tiles are computed TRANSPOSED (softmax/squash axis on the M/register
// axis) so reductions are in-lane; only one shfl_xor(16) per merge.
// Iteration 0 (Wacc==0 -> exactly uniform coupling) is strength-reduced to a
// column-sum + squash + broadcast; the WMMA path runs for iterations 1 and 2.
// ---------------------------------------------------------------------------

typedef float v2f __attribute__((ext_vector_type(2)));
typedef float v8f __attribute__((ext_vector_type(8)));

#define BATCH 32
#define RCAP  576          // routes == capsules
#define DDIM  16
#define CIN   8
#define NTILE 36           // 576 / 16
#define NWAVE 16           // 512 threads
#define USTR  18           // padded row stride for U/Wacc (floats): 18g mod 64 distinct
#define EST   596          // padded E-stage row stride (floats): 596 % 64 == 20

// LDS layout (bytes)
#define OFF_U     0
#define OFF_WACC  (RCAP*USTR*4)               // 41472
#define OFF_EST   (2*RCAP*USTR*4)             // 82944
#define OFF_M     (OFF_EST + 16*EST*4)        // 121088
#define OFF_RZ    (OFF_M + RCAP*4)            // 123392
#define LDS_BYTES (OFF_RZ + RCAP*4)           // 125696  (< 320 KB/WGP)

// ---------------------------------------------------------------------------
// Kernel 1: u_core[b,r,j] = sum_k x[b,k,r] * W[b,r,k,j]
// One thread per (b,r,j-quad); float4 W loads / u stores, streaming-bound.
// ---------------------------------------------------------------------------
__global__ void capsule_ucore_kernel(const float* __restrict__ x,
                                     const float* __restrict__ w,
                                     float* __restrict__ u) {
  int tid = blockIdx.x * blockDim.x + threadIdx.x;
  if (tid >= BATCH * RCAP * 4) return;
  int j4 = tid & 3;
  int br = tid >> 2;                 // b*R + r
  int r  = br % RCAP;
  int b  = br / RCAP;
  const float* xp = x + (size_t)(b * CIN) * RCAP + r;            // x[b][k][r]
  const float* wp = w + (size_t)br * (CIN * DDIM) + j4 * 4;      // W[b][r][k][j4*4]
  float4 acc = {0.f, 0.f, 0.f, 0.f};
#pragma unroll
  for (int k = 0; k < CIN; ++k) {
    float  xv = xp[k * RCAP];
    float4 wv = *(const float4*)&wp[k * DDIM];
    acc.x = fmaf(xv, wv.x, acc.x);
    acc.y = fmaf(xv, wv.y, acc.y);
    acc.z = fmaf(xv, wv.z, acc.z);
    acc.w = fmaf(xv, wv.w, acc.w);
  }
  *(float4*)&u[(size_t)br * DDIM + j4 * 4] = acc;
}

// Transposed logit tile: Lt[i0+mi, r0+g] = sum_k Wa[i0+mi][k] * U[r0+g][k].
// A (16x4 chunk): lane(g,half) reg v = Wa[(i0+g)*USTR + k0+2*half+v]  (float2)
// B (4x16 chunk): lane(g,half) reg v = U [(r0+g)*USTR + k0+2*half+v]  (float2)
// C: lane(g,half) holds rows i = i0 + half*8 + v, column r = r0 + g.
__device__ inline v8f logitT_tile(const float* __restrict__ Wa,
                                  const float* __restrict__ Us,
                                  int i0, int r0, int g, int half) {
  v8f L = {0.f, 0.f, 0.f, 0.f, 0.f, 0.f, 0.f, 0.f};
#pragma unroll
  for (int kc = 0; kc < 4; ++kc) {
    const int kb = kc * 4 + 2 * half;
    v2f a  = *(const v2f*)&Wa[(i0 + g) * USTR + kb];
    v2f bb = *(const v2f*)&Us[(r0 + g) * USTR + kb];
    L = __builtin_amdgcn_wmma_f32_16x16x4_f32(false, a, false, bb,
                                              (short)0, L, false, false);
  }
  return L;
}

// ---------------------------------------------------------------------------
// Kernel 2: full 3-iteration routing for one batch per workgroup (16 waves).
// ---------------------------------------------------------------------------
__launch_bounds__(512, 1)
__global__ void capsule_routing_kernel(const float* __restrict__ uin,
                                       float* __restrict__ out) {
  extern __shared__ char smem[];
  float* Us  = (float*)(smem + OFF_U);     // U      [576][USTR]
  float* Wa  = (float*)(smem + OFF_WACC);  // Wacc   [576][USTR]
  float* Es  = (float*)(smem + OFF_EST);   // E tile [16][EST] (+ t=0 scratch)
  float* Ms  = (float*)(smem + OFF_M);     // row max [576]
  float* RZs = (float*)(smem + OFF_RZ);    // 1/row-sumexp [576]

  const int b    = blockIdx.x;
  const int tid  = threadIdx.x;
  const int wave = tid >> 5;
  const int lane = tid & 31;
  const int g    = lane & 15;
  const int half = lane >> 4;

  // Load U (padded rows).
  const float* ug = uin + (size_t)b * RCAP * DDIM;
  for (int idx = tid; idx < RCAP * DDIM; idx += 512) {
    int row = idx >> 4, col = idx & 15;
    Us[row * USTR + col] = ug[idx];
  }
  __syncthreads();

  // ---- Iteration 0 fast path (exact): Wacc==0 -> logits are exactly 0,
  // coupling is exactly uniform, so v0[j] = squash(sum_r U[r][j] / 576),
  // identical for every capsule i. Wacc <- v0 broadcast.
  {
    int j = tid & 15, c = tid >> 4;                 // c in 0..31, 32*18 == 576
    float p = 0.0f;
    for (int r = c * 18; r < c * 18 + 18; ++r) p += Us[r * USTR + j];
    Es[c * 16 + j] = p;
    __syncthreads();
    if (tid < 16) {
      float s = 0.0f;
      for (int c2 = 0; c2 < 32; ++c2) s += Es[c2 * 16 + tid];
      Es[512 + tid] = s * (1.0f / 576.0f);          // sbar[j]
    }
    __syncthreads();
    if (tid == 0) {
      float nrm = 0.0f;
      for (int jj = 0; jj < 16; ++jj) { float sv = Es[512 + jj]; nrm += sv * sv; }
      Es[528] = (nrm / (1.0f + nrm)) * rsqrtf(nrm + 1e-8f);
    }
    __syncthreads();
    float scale = Es[528];
    for (int idx = tid; idx < RCAP * DDIM; idx += 512) {
      int row = idx >> 4, col = idx & 15;
      Wa[row * USTR + col] = Es[512 + col] * scale; // v0, same for all rows
    }
    __syncthreads();
  }

  // ---- Iterations 1 and 2: full WMMA routing ----
  for (int t = 1; t < 3; ++t) {
    // -- Phase 1: softmax stats per route r over capsule axis i.
    // Wave owns r-tile (N axis); loops i-tiles; stats stay in-lane.
    for (int rt = wave; rt < NTILE; rt += NWAVE) {
      const int r0 = rt * 16;
      float mrun = -3.0e38f, zrun = 0.0f;
      for (int it = 0; it < NTILE; ++it) {
        v8f L = logitT_tile(Wa, Us, it * 16, r0, g, half);
        float tmax = L[0];
#pragma unroll
        for (int v = 1; v < 8; ++v) tmax = fmaxf(tmax, L[v]);
        float mnew = fmaxf(mrun, tmax);
        float zs = 0.0f;
#pragma unroll
        for (int v = 0; v < 8; ++v) zs += __expf(L[v] - mnew);
        zrun = zrun * __expf(mrun - mnew) + zs;
        mrun = mnew;
      }
      // merge the two half-wave row groups: 2 shuffles total
      float om = __shfl_xor(mrun, 16, 32);
      float oz = __shfl_xor(zrun, 16, 32);
      float mt = fmaxf(mrun, om);
      float zt = zrun * __expf(mrun - mt) + oz * __expf(om - mt);
      if (half == 0) { Ms[r0 + g] = mt; RZs[r0 + g] = 1.0f / zt; }
    }
    __syncthreads();

    // -- Phase 2: S^T[j,i] = sum_r U[r][j] * c[r,i], tiled over r.
    v8f acc[3] = {};
    for (int rt = 0; rt < NTILE; ++rt) {
      const int r0 = rt * 16;
      // Produce E rows for this r-tile: c[r,i] = exp(Lt - m[r]) / z[r]
      for (int it = wave; it < NTILE; it += NWAVE) {
        const int i0 = it * 16;
        v8f L = logitT_tile(Wa, Us, i0, r0, g, half);
        const float m  = Ms[r0 + g];
        const float rz = RZs[r0 + g];
        float4 c0, c1;
        c0.x = __expf(L[0] - m) * rz;  c0.y = __expf(L[1] - m) * rz;
        c0.z = __expf(L[2] - m) * rz;  c0.w = __expf(L[3] - m) * rz;
        c1.x = __expf(L[4] - m) * rz;  c1.y = __expf(L[5] - m) * rz;
        c1.z = __expf(L[6] - m) * rz;  c1.w = __expf(L[7] - m) * rz;
        float* ep = &Es[g * EST + i0 + half * 8];
        *(float4*)ep       = c0;
        *(float4*)(ep + 4) = c1;
      }
      __syncthreads();
      // Consume: accT(i-tile) += U^T(16j x 16r) * E(16r x 16i)
      {
        int nt = 0;
        for (int it = wave; it < NTILE; it += NWAVE, ++nt) {
          const int i0 = it * 16;
          v8f A = acc[nt];
#pragma unroll
          for (int kc = 0; kc < 4; ++kc) {
            const int kb = kc * 4 + 2 * half;
            v2f a, bb;
            a.x  = Us[(r0 + kb) * USTR + g];       // A[j=g][k] = U[r0+k][j]
            a.y  = Us[(r0 + kb + 1) * USTR + g];
            bb.x = Es[kb * EST + i0 + g];          // B[k][i] = E[k][i]
            bb.y = Es[(kb + 1) * EST + i0 + g];
            A = __builtin_amdgcn_wmma_f32_16x16x4_f32(false, a, false, bb,
                                                      (short)0, A, false, false);
          }
          acc[nt] = A;
        }
      }
      __syncthreads();
    }

    // -- Phase 3: squash columns of S^T; update Wacc or emit output.
    {
      int nt = 0;
      for (int it = wave; it < NTILE; it += NWAVE, ++nt) {
        const int i = it * 16 + g;
        v8f S = acc[nt];
        float nh = 0.0f;
#pragma unroll
        for (int v = 0; v < 8; ++v) nh += S[v] * S[v];
        float nrm   = nh + __shfl_xor(nh, 16, 32);   // merge j halves
        float scale = (nrm / (1.0f + nrm)) * rsqrtf(nrm + 1e-8f);
        if (t < 2) {
          float* wp = &Wa[i * USTR + half * 8];      // 8 contiguous floats
#pragma unroll
          for (int v = 0; v < 8; ++v) wp[v] += S[v] * scale;
        } else {
          float* op = out + ((size_t)b * RCAP + i) * DDIM + half * 8;
          float4 o0, o1;
          o0.x = S[0] * scale; o0.y = S[1] * scale;
          o0.z = S[2] * scale; o0.w = S[3] * scale;
          o1.x = S[4] * scale; o1.y = S[5] * scale;
          o1.z = S[6] * scale; o1.w = S[7] * scale;
          *(float4*)op       = o0;
          *(float4*)(op + 4) = o1;
        }
      }
    }
    __syncthreads();
  }
}

// ---------------------------------------------------------------------------
extern "C" void kernel_launch(void* const* d_in, const int* in_sizes, int n_in,
                              void* d_out, int out_size, void* d_ws, size_t ws_size,
                              hipStream_t stream) {
  const float* x = (const float*)d_in[0];   // [32, 8, 24, 24]
  const float* w = (const float*)d_in[1];   // [32, 576, 8, 16]
  float* u   = (float*)d_ws;                // u_core scratch [32,576,16] = 1.18 MB
  float* out = (float*)d_out;               // v [32, 576, 16]

  const int n1 = BATCH * RCAP * 4;          // 73728 threads (float4 per thread)
  capsule_ucore_kernel<<<(n1 + 255) / 256, 256, 0, stream>>>(x, w, u);
  capsule_routing_kernel<<<BATCH, 512, LDS_BYTES, stream>>>(u, out);
}